// ConvCaps3D_16320875725239
// MI455X (gfx1250) — compile-verified
//
#include <hip/hip_runtime.h>
#include <hip/hip_bf16.h>

typedef __attribute__((ext_vector_type(2))) float v2f;
typedef __attribute__((ext_vector_type(8))) float v8f;

#define HW      94
#define PTS     8836        // 94*94
#define PT3     70688       // 8*94*94 points per batch (capdim x h x w)
#define KTOT    144         // 16 ic * 3 * 3
#define TILE_P  16
#define NTILES  553         // ceil(8836/16)

// ---------------------------------------------------------------------------
// Kernel 1: conv as im2col GEMM with FP32 WMMA (16x16x4), exact fp32 math.
// grid: x = point tile (553), y = B*8 + depth (64)
// block: 256 threads = 8 waves; wave w owns channel groups w*16 and 128+w*16,
// so one block covers all 256 output channels and stages im2col LDS once.
// u layout: u[(bz*256 + oc)*PTS + pt],  bz = B*8 + depth, oc = cap_c*8+cap_d
// ---------------------------------------------------------------------------
__global__ __launch_bounds__(256) void conv_wmma_kernel(
    const float* __restrict__ x, const float* __restrict__ w,
    float* __restrict__ u)
{
    __shared__ float bs[KTOT * TILE_P];   // im2col tile: 144 x 16 = 9 KB

    const int tile = blockIdx.x;
    const int bz   = blockIdx.y;          // B*8 + depth
    const int B    = bz >> 3;
    const int Dz   = bz & 7;
    const int p0   = tile * TILE_P;

    // ---- stage im2col tile into LDS (coalesced over pt) ----
    for (int e = threadIdx.x; e < KTOT * TILE_P; e += 256) {
        const int k  = e >> 4;
        const int pt = e & 15;
        const int p  = p0 + pt;
        float val = 0.0f;
        if (p < PTS) {
            const int ic  = k / 9;
            const int rem = k - ic * 9;
            const int r   = rem / 3;
            const int cc  = rem - r * 3;
            const int yy  = p / HW;
            const int xx  = p - yy * HW;
            val = x[((((size_t)B * 16 + ic) * 8 + Dz) * 96 + (yy + r)) * 96
                    + (xx + cc)];
        }
        bs[k * TILE_P + pt] = val;
    }
    __syncthreads();

    const int lane    = threadIdx.x & 31;
    const int wv      = threadIdx.x >> 5;         // 0..7
    const int chbase0 = wv * 16;                  // first 16-channel group
    const int chbase1 = 128 + wv * 16;            // second 16-channel group
    const int khalf   = lane >> 4;                // 0 or 1
    const int chl     = lane & 15;

    // ---- preload A fragments (weights): 36 K-steps of 4, two groups ----
    const float* wp0 = w + (size_t)(chbase0 + chl) * KTOT + 2 * khalf;
    const float* wp1 = w + (size_t)(chbase1 + chl) * KTOT + 2 * khalf;
    v2f areg0[36], areg1[36];
#pragma unroll
    for (int kk = 0; kk < 36; ++kk) {
        areg0[kk] = *(const v2f*)(wp0 + 4 * kk);  // 8B aligned (even fp32 offs)
        areg1[kk] = *(const v2f*)(wp1 + 4 * kk);
    }

    // ---- GEMM: 72 chained v_wmma_f32_16x16x4_f32 (B fragment reused x2) ----
    const int ptl = lane & 15;
    const int kh2 = khalf * 2;
    v8f acc0 = {}, acc1 = {};
#pragma unroll
    for (int kk = 0; kk < 36; ++kk) {
        v2f bfrag;
        bfrag.x = bs[(4 * kk + kh2)     * TILE_P + ptl];
        bfrag.y = bs[(4 * kk + 1 + kh2) * TILE_P + ptl];
        acc0 = __builtin_amdgcn_wmma_f32_16x16x4_f32(
                   false, areg0[kk], false, bfrag, (short)0, acc0, false, false);
        acc1 = __builtin_amdgcn_wmma_f32_16x16x4_f32(
                   false, areg1[kk], false, bfrag, (short)0, acc1, false, false);
    }

    // ---- store D: VGPR j -> channel row (j + 8*half), col = pt ----
    const int p = p0 + ptl;
    if (p < PTS) {
        const int rowadd = khalf * 8;
        const size_t base0 = ((size_t)bz * 256 + chbase0 + rowadd) * PTS + p;
        const size_t base1 = ((size_t)bz * 256 + chbase1 + rowadd) * PTS + p;
#pragma unroll
        for (int j = 0; j < 8; ++j) {
            u[base0 + (size_t)j * PTS] = acc0[j];
            u[base1 + (size_t)j * PTS] = acc1[j];
        }
    }
}

// ---------------------------------------------------------------------------
// Kernel 0: zero the softmax-denominator accumulators (Z2[64] | Z3[64])
// ---------------------------------------------------------------------------
__global__ void zero_z_kernel(float* __restrict__ Z)
{
    if (threadIdx.x < 128) Z[threadIdx.x] = 0.0f;
}

// ---------------------------------------------------------------------------
// Routing iteration (iters 1 and 2). One thread per point (B, pt3).
// Single streamed pass over the 32x8 u-slice accumulating the 8x8 Gram
// matrix G = U*U^T (36 regs): b_upd = scale*(G*ccoef), n2 = ccoef.(G*ccoef).
// ~80 VGPRs -> high occupancy, one HBM read of u per iteration.
// ---------------------------------------------------------------------------
__global__ __launch_bounds__(256) void route_iter_kernel(
    const float* __restrict__ u, float* __restrict__ bbuf,
    const float* __restrict__ Zin, float* __restrict__ Zout, int first)
{
    const int B    = blockIdx.y;
    const int lane = threadIdx.x & 31;
    const int pt3  = blockIdx.x * 256 + threadIdx.x;
    const bool valid = (pt3 < PT3);
    const int p3 = valid ? pt3 : 0;
    const int d  = p3 / PTS;
    const int pt = p3 - d * PTS;

    const float* ub = u + (((size_t)B * 8 * 256 + d) * PTS + pt);

    // coupling coefficients
    float ccoef[8], bold[8];
    if (first) {
#pragma unroll
        for (int dz = 0; dz < 8; ++dz) {
            ccoef[dz] = 1.0f / (float)PT3;   // softmax of zeros
            bold[dz]  = 0.0f;
        }
    } else {
#pragma unroll
        for (int dz = 0; dz < 8; ++dz) {
            bold[dz]  = bbuf[(B * 8 + dz) * PT3 + p3];
            ccoef[dz] = __expf(bold[dz]) / Zin[B * 8 + dz];
        }
    }

    // stream u slice once, accumulate upper-triangular Gram G[i][j] (i<=j)
    float G[8][8];
#pragma unroll
    for (int i = 0; i < 8; ++i)
#pragma unroll
        for (int j = i; j < 8; ++j) G[i][j] = 0.0f;

#pragma unroll
    for (int c = 0; c < 32; ++c) {
        float uv[8];
#pragma unroll
        for (int dz = 0; dz < 8; ++dz)
            uv[dz] = ub[((size_t)dz * 256 + c * 8) * PTS];
#pragma unroll
        for (int i = 0; i < 8; ++i)
#pragma unroll
            for (int j = i; j < 8; ++j) G[i][j] += uv[i] * uv[j];
    }

    // a = G * ccoef (using symmetry), n2 = ccoef . a
    float a[8];
    float n2 = 0.0f;
#pragma unroll
    for (int i = 0; i < 8; ++i) {
        float acc = 0.0f;
#pragma unroll
        for (int j = 0; j < 8; ++j)
            acc += (j >= i ? G[i][j] : G[j][i]) * ccoef[j];
        a[i] = acc;
        n2 += acc * ccoef[i];
    }
    const float scale = (n2 / (1.0f + n2)) * rsqrtf(n2 + 1e-8f);

    // b_new[dz] = b_old[dz] + scale * a[dz]; accumulate Z_next = sum exp(b_new)
#pragma unroll
    for (int dz = 0; dz < 8; ++dz) {
        const float bn = bold[dz] + a[dz] * scale;
        if (valid) bbuf[(B * 8 + dz) * PT3 + pt3] = bn;

        float e = valid ? __expf(bn) : 0.0f;
#pragma unroll
        for (int off = 16; off; off >>= 1) e += __shfl_down(e, off, 32);
        if (lane == 0) atomicAdd(&Zout[B * 8 + dz], e);
    }
}

// ---------------------------------------------------------------------------
// Final iteration: compute s only, streamed, write output (B, c, d, h, w).
// ---------------------------------------------------------------------------
__global__ __launch_bounds__(256) void route_final_kernel(
    const float* __restrict__ u, const float* __restrict__ bbuf,
    const float* __restrict__ Z, float* __restrict__ out)
{
    const int B   = blockIdx.y;
    const int pt3 = blockIdx.x * 256 + threadIdx.x;
    if (pt3 >= PT3) return;
    const int d  = pt3 / PTS;
    const int pt = pt3 - d * PTS;

    float ccoef[8];
#pragma unroll
    for (int dz = 0; dz < 8; ++dz)
        ccoef[dz] = __expf(bbuf[(B * 8 + dz) * PT3 + pt3]) / Z[B * 8 + dz];

    const float* ub = u + (((size_t)B * 8 * 256 + d) * PTS + pt);
#pragma unroll
    for (int c = 0; c < 32; ++c) {
        float acc = 0.0f;
#pragma unroll
        for (int dz = 0; dz < 8; ++dz)
            acc += ccoef[dz] * ub[((size_t)dz * 256 + c * 8) * PTS];
        out[(((size_t)B * 32 + c) * 8 + d) * PTS + pt] = acc;
    }
}

// ---------------------------------------------------------------------------
extern "C" void kernel_launch(void* const* d_in, const int* in_sizes, int n_in,
                              void* d_out, int out_size, void* d_ws, size_t ws_size,
                              hipStream_t stream)
{
    const float* x = (const float*)d_in[0];   // (8,16,8,96,96)
    const float* w = (const float*)d_in[1];   // (256,16,1,3,3)
    float* out = (float*)d_out;               // (8,32,8,94,94)

    // workspace layout
    float* u = (float*)d_ws;
    const size_t uElems = (size_t)64 * 256 * PTS;     // 144,769,024 fp32
    float* bbuf = u + uElems;
    const size_t bElems = (size_t)64 * PT3;           // 4,524,032 fp32
    float* Z2 = bbuf + bElems;                        // 64
    float* Z3 = Z2 + 64;                              // 64

    // zero Z accumulators (deterministic per launch)
    zero_z_kernel<<<1, 128, 0, stream>>>(Z2);

    // conv -> u  (one block covers all 256 channels of a point tile)
    conv_wmma_kernel<<<dim3(NTILES, 64), 256, 0, stream>>>(x, w, u);

    const int rblocks = (PT3 + 255) / 256;   // 277
    // routing iter 1 (uniform c), writes b and Z2
    route_iter_kernel<<<dim3(rblocks, 8), 256, 0, stream>>>(u, bbuf, Z3, Z2, 1);
    // routing iter 2, reads b/Z2, writes b and Z3
    route_iter_kernel<<<dim3(rblocks, 8), 256, 0, stream>>>(u, bbuf, Z2, Z3, 0);
    // routing iter 3: s -> output
    route_final_kernel<<<dim3(rblocks, 8), 256, 0, stream>>>(u, bbuf, Z3, out);
}